// Decoder_56349970923609
// MI455X (gfx1250) — compile-verified
//
#include <hip/hip_runtime.h>

// Viterbi decode: B=64, S=512, T=32.
// One wave32 per batch; async global->LDS tile pipeline (CDNA5 ASYNCcnt path);
// backpointers kept in LDS; in-kernel traceback.

#define B_   64
#define S_   512
#define T_   32
#define NBUF 6   // ring depth: 6 tiles * 8 async ops = 48 outstanding (< 63)

// Broadcast lane i's value of v to all lanes via v_readlane (SGPR operand).
__device__ __forceinline__ float bcast_f(float v, int i) {
  return __int_as_float(__builtin_amdgcn_readlane(__float_as_int(v), i));
}

// Issue one 4KB tile as 8 x global_load_async_to_lds_b128 (32 lanes x 16B each).
// VDST operand = per-lane LDS byte offset, VADDR = per-lane 64-bit global address.
__device__ __forceinline__ void async_tile(unsigned lds_off, const float* gbase, int lane) {
#pragma unroll
  for (int s = 0; s < 8; ++s) {
    unsigned      loff = lds_off + (unsigned)(s * 512 + lane * 16);
    const char*   g    = (const char*)gbase + s * 512 + lane * 16;
    asm volatile("global_load_async_to_lds_b128 %0, %1, off"
                 :: "v"(loff), "v"(g) : "memory");
  }
}

__global__ __launch_bounds__(T_) void viterbi_decode_kernel(
    const float* __restrict__ lp,          // [B,S,T,T]
    const unsigned char* __restrict__ mask, // [B,S] (bool)
    float* __restrict__ out)                // [B] scores, then [B,S] tags-as-float
{
  __shared__ float         ring[NBUF][T_ * T_]; // 24 KB tile ring
  __shared__ float         dummy[T_ * T_];      // 4 KB sink for tail cadence
  __shared__ unsigned char bp[S_ * T_];         // 16 KB backpointers

  const int    b    = blockIdx.x;
  const int    lane = threadIdx.x;              // wave32: lane == column j
  const float* base = lp + (size_t)b * S_ * T_ * T_;

  const unsigned ringOff  = (unsigned)(unsigned long long)(const void*)&ring[0][0];
  const unsigned dummyOff = (unsigned)(unsigned long long)(const void*)&dummy[0];

  // Prologue: fill the ring (48 async ops in flight).
  for (int t = 0; t < NBUF; ++t)
    async_tile(ringOff + (unsigned)(t * T_ * T_ * 4), base + (size_t)t * T_ * T_, lane);

  float alpha = 0.0f;
  for (int t = 0; t < S_; ++t) {
    // Issued so far: 48 + 8t.  Tiles 0..t done  <=>  ASYNCcnt <= 40 (in-order).
    asm volatile("s_wait_asynccnt 40" ::: "memory");
    const float* tl = &ring[t % NBUF][0];

    if (t == 0) {
      float best = tl[lane];                    // alpha0[j] = max_i lp[0,i,j]
#pragma unroll
      for (int i = 1; i < T_; ++i) best = fmaxf(best, tl[i * T_ + lane]);
      alpha = best;
    } else {
      float best = bcast_f(alpha, 0) + tl[lane];
      int   arg  = 0;
#pragma unroll
      for (int i = 1; i < T_; ++i) {
        float cand = bcast_f(alpha, i) + tl[i * T_ + lane];
        if (cand > best) { best = cand; arg = i; }   // strict '>' => first-max tie rule
      }
      alpha = best;
      bp[t * T_ + lane] = (unsigned char)arg;
    }

    // Keep exactly 8 async ops issued per step so the wait immediate stays valid.
    const int nt = t + NBUF;
    if (nt < S_)
      async_tile(ringOff + (unsigned)((nt % NBUF) * T_ * T_ * 4),
                 base + (size_t)nt * T_ * T_, lane);
    else
      async_tile(dummyOff, base + (size_t)(S_ - 1) * T_ * T_, lane);
  }

  // Final max / argmax over lanes (uniform via readlane chain).
  float best = bcast_f(alpha, 0);
  int   cur  = 0;
#pragma unroll
  for (int i = 1; i < T_; ++i) {
    float v = bcast_f(alpha, i);
    if (v > best) { best = v; cur = i; }
  }

  // Drain remaining (dummy) async ops.
  asm volatile("s_wait_asynccnt 0" ::: "memory");

  float* outTags = out + B_;
  const unsigned char* mrow = mask + (size_t)b * S_;
  if (lane == 0) {
    out[b] = best;
    outTags[(size_t)b * S_ + (S_ - 1)] = mrow[S_ - 1] ? (float)cur : -1.0f;
  }
  // Traceback from LDS: y_{t-1} = bp[t][y_t]  (uniform dependent ds loads).
  for (int t = S_ - 1; t >= 1; --t) {
    int prev = bp[t * T_ + cur];
    if (lane == 0)
      outTags[(size_t)b * S_ + (t - 1)] = mrow[t - 1] ? (float)prev : -1.0f;
    cur = prev;
  }
}

extern "C" void kernel_launch(void* const* d_in, const int* in_sizes, int n_in,
                              void* d_out, int out_size, void* d_ws, size_t ws_size,
                              hipStream_t stream) {
  (void)in_sizes; (void)n_in; (void)out_size; (void)d_ws; (void)ws_size;
  const float*         lp   = (const float*)d_in[0];
  const unsigned char* mask = (const unsigned char*)d_in[1];
  float*               out  = (float*)d_out;
  viterbi_decode_kernel<<<dim3(B_), dim3(T_), 0, stream>>>(lp, mask, out);
}